// GATRNN_74543452389747
// MI455X (gfx1250) — compile-verified
//
#include <hip/hip_runtime.h>
#include <cstdint>
#include <cstddef>

// Problem constants (match reference)
#define NN   20000
#define TT   24
#define FNS  32
#define FES  8
#define FND  16
#define HH_  128
#define EE   320000
#define G4   512           // 4*H gate outputs
#define NEG_SLOPE 0.2f

typedef __attribute__((ext_vector_type(2))) float v2f;
typedef __attribute__((ext_vector_type(8))) float v8f;

__device__ __forceinline__ v8f v8f_zero() {
    v8f z;
#pragma unroll
    for (int i = 0; i < 8; ++i) z[i] = 0.f;
    return z;
}

// Monotonic float<->uint encoding so unsigned atomicMax == float max (handles negatives).
__device__ __forceinline__ unsigned enc_f(float x) {
    unsigned u = __float_as_uint(x);
    return (u & 0x80000000u) ? ~u : (u | 0x80000000u);
}
__device__ __forceinline__ float dec_f(unsigned u) {
    unsigned b = (u & 0x80000000u) ? (u ^ 0x80000000u) : ~u;
    return __uint_as_float(b);
}

__device__ __forceinline__ float sigmoidf_(float x) {
    return 1.f / (1.f + __expf(-x));
}

// ---------------------------------------------------------------------------
// 0) init workspace accumulators
// ---------------------------------------------------------------------------
__global__ void init_kernel(float* __restrict__ accum, float* __restrict__ denom,
                            unsigned* __restrict__ amax) {
    int idx = blockIdx.x * blockDim.x + threadIdx.x;
    if (idx < NN * HH_) accum[idx] = 0.f;
    if (idx < NN) { denom[idx] = 0.f; amax[idx] = 0x007FFFFFu; /* enc(-inf) */ }
}

// ---------------------------------------------------------------------------
// 1) xl = x0 @ Wl^T + bl ; xr = x0 @ Wr^T + br   (x0 = node_static[:,0,:])
// ---------------------------------------------------------------------------
__global__ void node_transform_kernel(const float* __restrict__ ns,
                                      const float* __restrict__ Wl, const float* __restrict__ bl,
                                      const float* __restrict__ Wr, const float* __restrict__ br,
                                      float* __restrict__ xl, float* __restrict__ xr) {
    int idx = blockIdx.x * blockDim.x + threadIdx.x;
    if (idx >= NN * HH_) return;
    int n = idx >> 7, h = idx & 127;
    const float* x = ns + (size_t)n * (TT * FNS);   // t = 0 slice
    float sl = bl[h], sr = br[h];
#pragma unroll
    for (int k = 0; k < FNS; ++k) {
        float xv = x[k];
        sl += xv * Wl[h * FNS + k];
        sr += xv * Wr[h * FNS + k];
    }
    xl[idx] = sl;
    xr[idx] = sr;
}

// ---------------------------------------------------------------------------
// 2) per-edge attention logit a[e] = leaky_relu(xl[src]+xr[dst]+ea@We^T) . att
// ---------------------------------------------------------------------------
__global__ void edge_logits_kernel(const float* __restrict__ es, const int* __restrict__ ei,
                                   const float* __restrict__ xl, const float* __restrict__ xr,
                                   const float* __restrict__ We, const float* __restrict__ att,
                                   float* __restrict__ abuf, unsigned* __restrict__ amax) {
    int e = blockIdx.x;
    int h = threadIdx.x;                       // 128 threads
    int src = ei[e];
    int dst = ei[EE + e];
    const float* ea = es + (size_t)e * (TT * FES);   // t = 0 slice
    float m = xl[src * HH_ + h] + xr[dst * HH_ + h];
#pragma unroll
    for (int k = 0; k < FES; ++k) m += ea[k] * We[h * FES + k];
    m = (m > 0.f) ? m : NEG_SLOPE * m;
    float v = m * att[h];
#pragma unroll
    for (int off = 16; off > 0; off >>= 1) v += __shfl_xor(v, off, 32);
    __shared__ float s[4];
    if ((h & 31) == 0) s[h >> 5] = v;
    __syncthreads();
    if (h == 0) {
        float a = s[0] + s[1] + s[2] + s[3];
        abuf[e] = a;
        atomicMax(&amax[dst], enc_f(a));
    }
}

// ---------------------------------------------------------------------------
// 3) ex[e] = exp(a - amax[dst]); denom[dst] += ex
// ---------------------------------------------------------------------------
__global__ void edge_softmax_kernel(const float* __restrict__ abuf, const int* __restrict__ ei,
                                    const unsigned* __restrict__ amax,
                                    float* __restrict__ exbuf, float* __restrict__ denom) {
    int e = blockIdx.x * blockDim.x + threadIdx.x;
    if (e >= EE) return;
    int dst = ei[EE + e];
    float ex = __expf(abuf[e] - dec_f(amax[dst]));
    exbuf[e] = ex;
    atomicAdd(&denom[dst], ex);
}

// ---------------------------------------------------------------------------
// 4) accum[dst,h] += ex[e] * xl[src,h]
// ---------------------------------------------------------------------------
__global__ void edge_aggregate_kernel(const float* __restrict__ exbuf, const int* __restrict__ ei,
                                      const float* __restrict__ xl, float* __restrict__ accum) {
    long long idx = (long long)blockIdx.x * blockDim.x + threadIdx.x;
    if (idx >= (long long)EE * HH_) return;
    int e = (int)(idx >> 7);
    int h = (int)(idx & 127);
    int src = ei[e];
    int dst = ei[EE + e];
    atomicAdd(&accum[dst * HH_ + h], exbuf[e] * xl[src * HH_ + h]);
}

// ---------------------------------------------------------------------------
// 5) gat = relu(accum/denom + b_gat)  (in place; empty segments -> relu(b_gat))
// ---------------------------------------------------------------------------
__global__ void gat_finalize_kernel(float* __restrict__ accum, const float* __restrict__ denom,
                                    const float* __restrict__ b_gat) {
    int idx = blockIdx.x * blockDim.x + threadIdx.x;
    if (idx >= NN * HH_) return;
    int n = idx >> 7, h = idx & 127;
    float d = denom[n];
    float g = (d > 0.f) ? (accum[idx] / d) : 0.f;
    g += b_gat[h];
    accum[idx] = (g > 0.f) ? g : 0.f;
}

// ---------------------------------------------------------------------------
// 6) Time-invariant LSTM pre-activation as a tiled WMMA GEMM (software-
//    pipelined K loop, hoisted base pointers). Output in fragment-blocked
//    layout preF[block][wave][frag][lane][8].
// LDS floats: wihP 64*512*2 | gbuf 32*130
// ---------------------------------------------------------------------------
#define PRE_SMEM_FLOATS (64 * G4 * 2 + 32 * 130)

__global__ void pre_gemm_kernel(const float* __restrict__ gat, const float* __restrict__ Wih,
                                const float* __restrict__ bih, const float* __restrict__ bhh,
                                float* __restrict__ preF) {
    extern __shared__ float smem[];
    float* wihP = smem;                     // ((k>>1)*512 + o)*2 + (k&1), k<128
    float* gbuf = wihP + 64 * G4 * 2;       // [32][130] padded rows

    const int tid = threadIdx.x;
    const int n0 = blockIdx.x * 32;

    for (int idx = tid; idx < G4 * HH_; idx += 256) {
        int o = idx >> 7, k = idx & 127;
        wihP[(((k >> 1) * G4) + o) * 2 + (k & 1)] = Wih[o * (HH_ + FND) + k];
    }
    for (int idx = tid; idx < 32 * HH_; idx += 256) {
        int i = idx >> 7, k = idx & 127;
        gbuf[i * 130 + k] = gat[(size_t)(n0 + i) * HH_ + k];
    }
    __syncthreads();

    const int wave = tid >> 5;
    const int lane = tid & 31;
    const int nt = wave >> 2;
    const int colbase = (wave & 3) * 32;
    const int nsub = lane & 15;
    const int koff  = (lane < 16) ? 0 : 2;
    const int kpoff = (lane < 16) ? 0 : 1;
    const int arow  = nt * 16 + nsub;

    // Hoisted, loop-invariant base pointers; in-loop address = base + kb*stride
    const float* baseA = &gbuf[arow * 130 + koff];
    const float* baseB[8];
#pragma unroll
    for (int f = 0; f < 8; ++f) {
        int col = (f >> 1) * HH_ + colbase + (f & 1) * 16 + nsub;
        baseB[f] = &wihP[(kpoff * G4 + col) * 2];
    }
    auto loadA = [&](int kb) -> v2f {
        return *reinterpret_cast<const v2f*>(baseA + kb);
    };
    auto loadB = [&](int kb, int f) -> v2f {
        return *reinterpret_cast<const v2f*>(baseB[f] + kb * G4);  // kp=kb/2 -> kb*G4 floats
    };

    v8f acc[8];
#pragma unroll
    for (int f = 0; f < 8; ++f) {
        int col = (f >> 1) * HH_ + colbase + (f & 1) * 16 + nsub;
        float b0 = bih[col] + bhh[col];
        v8f t;
#pragma unroll
        for (int v = 0; v < 8; ++v) t[v] = b0;
        acc[f] = t;
    }

    // software-pipelined K loop
    v2f a_c = loadA(0);
    v2f b_c[8];
#pragma unroll
    for (int f = 0; f < 8; ++f) b_c[f] = loadB(0, f);
#pragma unroll 4
    for (int kb = 0; kb < HH_ - 4; kb += 4) {
        v2f a_n = loadA(kb + 4);
        v2f b_n[8];
#pragma unroll
        for (int f = 0; f < 8; ++f) b_n[f] = loadB(kb + 4, f);
#pragma unroll
        for (int f = 0; f < 8; ++f)
            acc[f] = __builtin_amdgcn_wmma_f32_16x16x4_f32(
                false, a_c, false, b_c[f], (short)0, acc[f], false, false);
        a_c = a_n;
#pragma unroll
        for (int f = 0; f < 8; ++f) b_c[f] = b_n[f];
    }
#pragma unroll
    for (int f = 0; f < 8; ++f)
        acc[f] = __builtin_amdgcn_wmma_f32_16x16x4_f32(
            false, a_c, false, b_c[f], (short)0, acc[f], false, false);

    // store fragments (32B contiguous per lane)
#pragma unroll
    for (int f = 0; f < 8; ++f) {
        size_t base = ((((size_t)blockIdx.x * 8 + wave) * 8 + f) * 32 + lane) * 8;
        *reinterpret_cast<v8f*>(&preF[base]) = acc[f];
    }
}

// ---------------------------------------------------------------------------
// 7) LSTM: 32 nodes/block, 8 waves. Wave w: node-tile (w>>2), 32 gate-cols
//    (w&3)*32 across all 4 gates -> i/f/g/o for a (node,hdim) share a lane.
//    Accumulators re-initialized each step from preF; K loops software-
//    pipelined with hoisted per-fragment base pointers (independent address
//    induction chains -> loads can clause and waits can stage).
// LDS floats: whhP 64*512*2 | wdP 8*512*2 | hbuf 32*130 | nbuf 32*18 | red 32
// ---------------------------------------------------------------------------
#define LSTM_SMEM_FLOATS (64 * G4 * 2 + 8 * G4 * 2 + 32 * 130 + 32 * 18 + 32)

__global__ void lstm_kernel(const float* __restrict__ preF, const float* __restrict__ ndf,
                            const float* __restrict__ Whh, const float* __restrict__ Wih,
                            const float* __restrict__ Wfc, const float* __restrict__ bfc,
                            float* __restrict__ out) {
    extern __shared__ float smem[];
    float* whhP = smem;                       // ((k>>1)*512 + o)*2 + (k&1)
    float* wdP  = whhP + 64 * G4 * 2;
    float* hbuf = wdP + 8 * G4 * 2;           // [32][130] padded rows
    float* nbuf = hbuf + 32 * 130;            // [32][18]
    float* red  = nbuf + 32 * 18;             // [32]

    const int tid = threadIdx.x;
    const int n0 = blockIdx.x * 32;

    for (int idx = tid; idx < G4 * HH_; idx += 256) {
        int o = idx >> 7, k = idx & 127;
        whhP[(((k >> 1) * G4) + o) * 2 + (k & 1)] = Whh[idx];
    }
    for (int idx = tid; idx < G4 * FND; idx += 256) {
        int o = idx >> 4, k = idx & 15;
        wdP[(((k >> 1) * G4) + o) * 2 + (k & 1)] = Wih[o * (HH_ + FND) + HH_ + k];
    }
    for (int idx = tid; idx < 32 * 130; idx += 256) hbuf[idx] = 0.f;
    if (tid < 32) red[tid] = 0.f;

    const int wave = tid >> 5;
    const int lane = tid & 31;
    const int nt = wave >> 2;                 // node tile 0/1
    const int colbase = (wave & 3) * 32;      // 32 hdims per wave
    const int nsub = lane & 15;
    const int koff  = (lane < 16) ? 0 : 2;    // A/B K-offset per lane half
    const int kpoff = (lane < 16) ? 0 : 1;    // K-pair offset
    const int arow  = nt * 16 + nsub;         // A fragment row (node)
    const int nodeb = nt * 16 + ((lane < 16) ? 0 : 8);  // C fragment node base

    const size_t pbase = (((size_t)blockIdx.x * 8 + wave) * 8) * 256 + (size_t)lane * 8;

    // Hoisted, loop-invariant base pointers
    const float* baseAn = &nbuf[arow * 18 + koff];
    const float* baseAh = &hbuf[arow * 130 + koff];
    const float* baseBd[8];
    const float* baseBh[8];
#pragma unroll
    for (int f = 0; f < 8; ++f) {
        int col = (f >> 1) * HH_ + colbase + (f & 1) * 16 + nsub;
        baseBd[f] = &wdP[(kpoff * G4 + col) * 2];
        baseBh[f] = &whhP[(kpoff * G4 + col) * 2];
    }
    auto loadA_n = [&](int kb) -> v2f {
        return *reinterpret_cast<const v2f*>(baseAn + kb);
    };
    auto loadA_h = [&](int kb) -> v2f {
        return *reinterpret_cast<const v2f*>(baseAh + kb);
    };
    auto loadB_d = [&](int kb, int f) -> v2f {
        return *reinterpret_cast<const v2f*>(baseBd[f] + kb * G4);
    };
    auto loadB_h = [&](int kb, int f) -> v2f {
        return *reinterpret_cast<const v2f*>(baseBh[f] + kb * G4);
    };

    v8f cst[2]  = { v8f_zero(), v8f_zero() };   // cell state, in registers
    v8f hsum[2] = { v8f_zero(), v8f_zero() };   // sum_t h for final mean/FC

    __syncthreads();

#pragma unroll 1
    for (int t = 0; t < TT; ++t) {
        // stage dynamic features for this timestep
        for (int idx = tid; idx < 32 * FND; idx += 256) {
            int i = idx >> 4, j = idx & 15;
            nbuf[i * 18 + j] = ndf[(size_t)(n0 + i) * (TT * FND) + t * FND + j];
        }
        __syncthreads();

        // init accumulators from fragment-blocked pre (coalesced v8f loads)
        v8f acc[8];
#pragma unroll
        for (int f = 0; f < 8; ++f)
            acc[f] = *reinterpret_cast<const v8f*>(&preF[pbase + (size_t)f * 256]);

        // ---- dynamic-input contribution, K = 16 (pipelined) ----
        v2f a_c = loadA_n(0);
        v2f b_c[8];
#pragma unroll
        for (int f = 0; f < 8; ++f) b_c[f] = loadB_d(0, f);
#pragma unroll
        for (int kb = 0; kb < FND - 4; kb += 4) {
            v2f a_n = loadA_n(kb + 4);
            v2f b_n[8];
#pragma unroll
            for (int f = 0; f < 8; ++f) b_n[f] = loadB_d(kb + 4, f);
#pragma unroll
            for (int f = 0; f < 8; ++f)
                acc[f] = __builtin_amdgcn_wmma_f32_16x16x4_f32(
                    false, a_c, false, b_c[f], (short)0, acc[f], false, false);
            a_c = a_n;
#pragma unroll
            for (int f = 0; f < 8; ++f) b_c[f] = b_n[f];
        }
        // recurrent k-step 0 prologue issues under the last dyn WMMAs
        {
            v2f a_n = loadA_h(0);
            v2f b_n[8];
#pragma unroll
            for (int f = 0; f < 8; ++f) b_n[f] = loadB_h(0, f);
#pragma unroll
            for (int f = 0; f < 8; ++f)
                acc[f] = __builtin_amdgcn_wmma_f32_16x16x4_f32(
                    false, a_c, false, b_c[f], (short)0, acc[f], false, false);
            a_c = a_n;
#pragma unroll
            for (int f = 0; f < 8; ++f) b_c[f] = b_n[f];
        }
        // ---- recurrent contribution, K = 128 (pipelined) ----
#pragma unroll 4
        for (int kb = 0; kb < HH_ - 4; kb += 4) {
            v2f a_n = loadA_h(kb + 4);
            v2f b_n[8];
#pragma unroll
            for (int f = 0; f < 8; ++f) b_n[f] = loadB_h(kb + 4, f);
#pragma unroll
            for (int f = 0; f < 8; ++f)
                acc[f] = __builtin_amdgcn_wmma_f32_16x16x4_f32(
                    false, a_c, false, b_c[f], (short)0, acc[f], false, false);
            a_c = a_n;
#pragma unroll
            for (int f = 0; f < 8; ++f) b_c[f] = b_n[f];
        }
#pragma unroll
        for (int f = 0; f < 8; ++f)
            acc[f] = __builtin_amdgcn_wmma_f32_16x16x4_f32(
                false, a_c, false, b_c[f], (short)0, acc[f], false, false);

        __syncthreads();   // all hbuf reads complete before overwrite

        // gate nonlinearity + state update (i,f,g,o co-located per lane/slot)
#pragma unroll
        for (int hh = 0; hh < 2; ++hh) {
#pragma unroll
            for (int v = 0; v < 8; ++v) {
                float gi_ = sigmoidf_(acc[0 * 2 + hh][v]);
                float gf_ = sigmoidf_(acc[1 * 2 + hh][v]);
                float gg_ = tanhf(acc[2 * 2 + hh][v]);
                float go_ = sigmoidf_(acc[3 * 2 + hh][v]);
                float cn = gf_ * cst[hh][v] + gi_ * gg_;
                cst[hh][v] = cn;
                float hv = go_ * tanhf(cn);
                hsum[hh][v] += hv;
                hbuf[(nodeb + v) * 130 + colbase + hh * 16 + nsub] = hv;
            }
        }
        // next iteration's leading barrier orders these writes vs. reads
    }

    __syncthreads();
    // out[n] = (sum_t h) . Wfc / T + bfc   (LDS float atomics per node)
#pragma unroll
    for (int hh = 0; hh < 2; ++hh) {
        float wv = Wfc[colbase + hh * 16 + nsub];
#pragma unroll
        for (int v = 0; v < 8; ++v)
            atomicAdd(&red[nodeb + v], hsum[hh][v] * wv);
    }
    __syncthreads();
    if (tid < 32) out[n0 + tid] = red[tid] * (1.f / TT) + bfc[0];
}

// ---------------------------------------------------------------------------
extern "C" void kernel_launch(void* const* d_in, const int* in_sizes, int n_in,
                              void* d_out, int out_size, void* d_ws, size_t ws_size,
                              hipStream_t stream) {
    const float* ns    = (const float*)d_in[0];
    const float* es    = (const float*)d_in[1];
    const float* ndf   = (const float*)d_in[2];
    const int*   ei    = (const int*)d_in[3];
    const float* Wl    = (const float*)d_in[4];
    const float* bl    = (const float*)d_in[5];
    const float* Wr    = (const float*)d_in[6];
    const float* br    = (const float*)d_in[7];
    const float* We    = (const float*)d_in[8];
    const float* att   = (const float*)d_in[9];
    const float* b_gat = (const float*)d_in[10];
    const float* Wih   = (const float*)d_in[11];
    const float* Whh   = (const float*)d_in[12];
    const float* bih   = (const float*)d_in[13];
    const float* bhh   = (const float*)d_in[14];
    const float* Wfc   = (const float*)d_in[15];
    const float* bfc   = (const float*)d_in[16];
    float* out = (float*)d_out;

    // workspace layout (floats)
    float* ws = (float*)d_ws;
    float*    xl    = ws;                         // N*H
    float*    xr    = xl + (size_t)NN * HH_;      // N*H
    float*    abuf  = xr + (size_t)NN * HH_;      // E
    float*    exbuf = abuf + EE;                  // E
    unsigned* amax  = (unsigned*)(exbuf + EE);    // N
    float*    denom = (float*)(amax + NN);        // N
    float*    accum = denom + NN;                 // N*H (becomes gat)
    float*    preF  = accum + (size_t)NN * HH_;   // N*512 (fragment-blocked)

    (void)in_sizes; (void)n_in; (void)out_size; (void)ws_size;

    init_kernel<<<(NN * HH_ + 255) / 256, 256, 0, stream>>>(accum, denom, amax);
    node_transform_kernel<<<(NN * HH_ + 255) / 256, 256, 0, stream>>>(
        ns, Wl, bl, Wr, br, xl, xr);
    edge_logits_kernel<<<EE, 128, 0, stream>>>(es, ei, xl, xr, We, att, abuf, amax);
    edge_softmax_kernel<<<(EE + 255) / 256, 256, 0, stream>>>(abuf, ei, amax, exbuf, denom);
    edge_aggregate_kernel<<<(int)(((long long)EE * HH_ + 255) / 256), 256, 0, stream>>>(
        exbuf, ei, xl, accum);
    gat_finalize_kernel<<<(NN * HH_ + 255) / 256, 256, 0, stream>>>(accum, denom, b_gat);
    pre_gemm_kernel<<<NN / 32, 256, PRE_SMEM_FLOATS * sizeof(float), stream>>>(
        accum, Wih, bih, bhh, preF);
    lstm_kernel<<<NN / 32, 256, LSTM_SMEM_FLOATS * sizeof(float), stream>>>(
        preF, ndf, Whh, Wih, Wfc, bfc, out);
}